// EnhancedPGAT_CrossAttn_Layer_52561809769177
// MI455X (gfx1250) — compile-verified
//
#include <hip/hip_runtime.h>
#include <math.h>

#define DF 128

typedef float v2f __attribute__((ext_vector_type(2)));
typedef float v8f __attribute__((ext_vector_type(8)));

// ---------------------------------------------------------------- utilities
__global__ void fillk(float* __restrict__ p, float v, size_t n) {
  size_t i = (size_t)blockIdx.x * blockDim.x + threadIdx.x;
  if (i < n) p[i] = v;
}

__device__ __forceinline__ void atomicMaxF(float* addr, float val) {
  int* ai = (int*)addr;
  int cur = __float_as_int(*addr);
  while (__int_as_float(cur) < val) {
    int old = atomicCAS(ai, cur, __float_as_int(val));
    if (old == cur) break;
    cur = old;
  }
}

// ------------------------------------------------- fp32 WMMA GEMM: Y = X@W^T
// X: [M,128] row-major.  W rows: W[n*ldw + k], k in [0,128).
// BMODE: 0 = no bias, 1 = +bias[n], 2 = +rowscale[m]*bias[n].
// One wave computes a 32x16 tile (two 16x16 accumulators sharing each
// B fragment): per K-step, 3x b64 loads feed 2x V_WMMA_F32_16X16X4_F32.
template <int BMODE>
__global__ void gemm_xwt(const float* __restrict__ X, const float* __restrict__ Wp,
                         const float* __restrict__ bias, const float* __restrict__ rowscale,
                         float* __restrict__ Y, int M, int ldw) {
  const int lane = threadIdx.x & 31;
  const int wave = threadIdx.x >> 5;   // n-tile 0..7 (covers N=128)
  const int half = lane >> 4;          // 0: lanes 0-15, 1: lanes 16-31
  const int l16  = lane & 15;
  const int m0 = blockIdx.x * 32;      // two m-tiles: m0, m0+16
  const int n0 = wave * 16;
  int mA0 = m0 + l16;      if (mA0 >= M) mA0 = M - 1;   // clamp; stores guarded
  int mA1 = m0 + 16 + l16; if (mA1 >= M) mA1 = M - 1;
  const float* xrow0 = X + (size_t)mA0 * DF;
  const float* xrow1 = X + (size_t)mA1 * DF;
  const float* wrow  = Wp + (size_t)(n0 + l16) * ldw;
  v8f acc0 = {0.f,0.f,0.f,0.f,0.f,0.f,0.f,0.f};
  v8f acc1 = {0.f,0.f,0.f,0.f,0.f,0.f,0.f,0.f};
#pragma unroll 8
  for (int k0 = 0; k0 < DF; k0 += 4) {
    const int ka = k0 + 2 * half;                 // per-half K pair (ISA A/B layout)
    v2f b  = *(const v2f*)(wrow  + ka);           // B[ka][n] = W[n][ka..ka+1]
    v2f a0 = *(const v2f*)(xrow0 + ka);
    v2f a1 = *(const v2f*)(xrow1 + ka);
    acc0 = __builtin_amdgcn_wmma_f32_16x16x4_f32(false, a0, false, b, (short)0, acc0,
                                                 false, false);
    acc1 = __builtin_amdgcn_wmma_f32_16x16x4_f32(false, a1, false, b, (short)0, acc1,
                                                 false, false);
  }
  const int n = n0 + l16;
  const float badd = (BMODE != 0) ? bias[n] : 0.f;
#pragma unroll
  for (int v = 0; v < 8; ++v) {
    const int ma = m0 + v + 8 * half;             // C/D: M = vgpr + 8*half
    const int mb = ma + 16;
    if (ma < M) {
      float r = acc0[v];
      if (BMODE == 1) r += badd;
      if (BMODE == 2) r += rowscale[ma] * badd;
      Y[(size_t)ma * DF + n] = r;
    }
    if (mb < M) {
      float r = acc1[v];
      if (BMODE == 1) r += badd;
      if (BMODE == 2) r += rowscale[mb] * badd;
      Y[(size_t)mb * DF + n] = r;
    }
  }
}

// -------------------------------------------- edge pass 1: comb & raw scores
// block = 128 threads = one edge; warp h owns head h (features 32h..32h+31)
__global__ void edge_pass1(const int* __restrict__ sidx, const int* __restrict__ tidx,
                           const float* __restrict__ A1, const float* __restrict__ B1,
                           const float* __restrict__ QE, const float* __restrict__ KE,
                           const float* __restrict__ Qm, const float* __restrict__ Km,
                           const float* __restrict__ m2w, const float* __restrict__ m2b,
                           const float* __restrict__ swp, const float* __restrict__ fwp,
                           float* __restrict__ comb, float* __restrict__ scores) {
  const int e = blockIdx.x;
  const int f = threadIdx.x;
  const int s = sidx[e], t = tidx[e];
  __shared__ float red[128];
  __shared__ float acc[12];   // [0..3]=attn qk, [4..7]=score qk, [8..11]=mlp2
  const size_t so = (size_t)s * DF + f;
  const size_t to = (size_t)t * DF + f;
  const float h  = fmaxf(A1[so] + B1[to], 0.f);      // relu(mlp1)
  const float qk = QE[so] * KE[to];                  // edge-weight attention
  const float sc = Qm[to] * Km[so];                  // conv scores (q from tgt)

  red[f] = qk; __syncthreads();
  if ((f & 31) < 16) red[f] += red[f + 16]; __syncthreads();
  if ((f & 31) < 8)  red[f] += red[f + 8];  __syncthreads();
  if ((f & 31) < 4)  red[f] += red[f + 4];  __syncthreads();
  if ((f & 31) < 2)  red[f] += red[f + 2];  __syncthreads();
  if ((f & 31) == 0) acc[f >> 5] = red[f] + red[f + 1];
  __syncthreads();

  red[f] = sc; __syncthreads();
  if ((f & 31) < 16) red[f] += red[f + 16]; __syncthreads();
  if ((f & 31) < 8)  red[f] += red[f + 8];  __syncthreads();
  if ((f & 31) < 4)  red[f] += red[f + 4];  __syncthreads();
  if ((f & 31) < 2)  red[f] += red[f + 2];  __syncthreads();
  if ((f & 31) == 0) acc[4 + (f >> 5)] = red[f] + red[f + 1];
  __syncthreads();

  for (int j = 0; j < 4; ++j) {                      // mlp2: 4 dot-128
    red[f] = h * m2w[j * DF + f]; __syncthreads();
    if (f < 64) red[f] += red[f + 64]; __syncthreads();
    if (f < 32) red[f] += red[f + 32]; __syncthreads();
    if (f < 16) red[f] += red[f + 16]; __syncthreads();
    if (f < 8)  red[f] += red[f + 8];  __syncthreads();
    if (f < 4)  red[f] += red[f + 4];  __syncthreads();
    if (f < 2)  red[f] += red[f + 2];  __syncthreads();
    if (f == 0) acc[8 + j] = red[0] + red[1];
    __syncthreads();
  }

  if (f < 4) {
    const float inv = 0.17677669529663687f;          // 1/sqrt(32)
    const float sw = swp[0], fw = fwp[0];
    const float mlpw = 1.f / (1.f + expf(-(acc[8 + f] + m2b[f])));
    const float attn = 1.f / (1.f + expf(-(acc[f] * inv)));
    comb[(size_t)e * 4 + f]   = sw * mlpw + fw * attn;
    scores[(size_t)e * 4 + f] = acc[4 + f] * inv;
  }
}

// ------------------------------- global (all-edge) softmax: max, exp+sum
__global__ void reduce_max4(const float* __restrict__ comb, float* __restrict__ gmax,
                            int total) {
  __shared__ float red[256];
  const int tid = threadIdx.x;                       // i mod 4 == tid mod 4
  float m = -INFINITY;
  for (int i = blockIdx.x * 256 + tid; i < total; i += gridDim.x * 256)
    m = fmaxf(m, comb[i]);
  red[tid] = m; __syncthreads();
  if (tid < 128) red[tid] = fmaxf(red[tid], red[tid + 128]); __syncthreads();
  if (tid < 64)  red[tid] = fmaxf(red[tid], red[tid + 64]);  __syncthreads();
  if (tid < 32)  red[tid] = fmaxf(red[tid], red[tid + 32]);  __syncthreads();
  if (tid < 16)  red[tid] = fmaxf(red[tid], red[tid + 16]);  __syncthreads();
  if (tid < 8)   red[tid] = fmaxf(red[tid], red[tid + 8]);   __syncthreads();
  if (tid < 4)   atomicMaxF(&gmax[tid], fmaxf(red[tid], red[tid + 4]));
}

__global__ void edge_exp_sum(float* __restrict__ buf, const float* __restrict__ gmax,
                             float* __restrict__ gsum, int total) {
  __shared__ float red[256];
  const int tid = threadIdx.x;
  const float m = gmax[tid & 3];
  float s = 0.f;
  for (int i = blockIdx.x * 256 + tid; i < total; i += gridDim.x * 256) {
    const float v = expf(buf[i] - m);
    buf[i] = v;
    s += v;
  }
  red[tid] = s; __syncthreads();
  if (tid < 128) red[tid] += red[tid + 128]; __syncthreads();
  if (tid < 64)  red[tid] += red[tid + 64];  __syncthreads();
  if (tid < 32)  red[tid] += red[tid + 32];  __syncthreads();
  if (tid < 16)  red[tid] += red[tid + 16];  __syncthreads();
  if (tid < 8)   red[tid] += red[tid + 8];   __syncthreads();
  if (tid < 4)   atomicAdd(&gsum[tid], red[tid] + red[tid + 4]);
}

// ------------------------- segment softmax over target neighborhoods
__global__ void edge_weighted(const int* __restrict__ tidx, const float* __restrict__ gsum,
                              const float* __restrict__ ewn, float* __restrict__ sc_w,
                              float* __restrict__ nmax, int total) {
  const int i = blockIdx.x * blockDim.x + threadIdx.x;
  if (i >= total) return;
  const int e = i >> 2, j = i & 3;
  const float w = sc_w[i] * (ewn[i] / gsum[j]);      // scores * edge_weight
  sc_w[i] = w;
  atomicMaxF(&nmax[(size_t)tidx[e] * 4 + j], w);
}

__global__ void edge_expnode(const int* __restrict__ tidx, const float* __restrict__ nmax,
                             const float* __restrict__ sc_w, float* __restrict__ pbuf,
                             float* __restrict__ nden, int total) {
  const int i = blockIdx.x * blockDim.x + threadIdx.x;
  if (i >= total) return;
  const int e = i >> 2, j = i & 3;
  const size_t idx = (size_t)tidx[e] * 4 + j;
  const float p = expf(sc_w[i] - nmax[idx]);
  pbuf[i] = p;
  atomicAdd(&nden[idx], p);
}

// ---------------------- scatter a*v and raw t_src rows into target nodes
__global__ void edge_scatter(const int* __restrict__ sidx, const int* __restrict__ tidx,
                             const float* __restrict__ pbuf, const float* __restrict__ nden,
                             const float* __restrict__ V, const float* __restrict__ Traw,
                             float* __restrict__ wv, float* __restrict__ ts,
                             float* __restrict__ deg) {
  const int e = blockIdx.x, f = threadIdx.x;
  const int s = sidx[e], t = tidx[e];
  const int j = f >> 5;
  const float a = pbuf[(size_t)e * 4 + j] / fmaxf(nden[(size_t)t * 4 + j], 1e-16f);
  atomicAdd(&wv[(size_t)t * DF + f], a * V[(size_t)s * DF + f]);
  atomicAdd(&ts[(size_t)t * DF + f], Traw[(size_t)s * DF + f]);
  if (f == 0) atomicAdd(&deg[t], 1.f);
}

// ---------------------------------- residual + relu + layernorm epilogue
__global__ void epilogue_ln(const float* __restrict__ prev, const float* __restrict__ upd,
                            const float* __restrict__ rwp, const float* __restrict__ eap,
                            const float* __restrict__ gamma, const float* __restrict__ beta,
                            float* __restrict__ dst) {
  const int r = blockIdx.x, f = threadIdx.x;
  __shared__ float red[128];
  const float u = upd[(size_t)r * DF + f];
  const float v = rwp[0] * prev[(size_t)r * DF + f] + eap[0] * fmaxf(u, 0.f);
  red[f] = v; __syncthreads();
  if (f < 64) red[f] += red[f + 64]; __syncthreads();
  if (f < 32) red[f] += red[f + 32]; __syncthreads();
  if (f < 16) red[f] += red[f + 16]; __syncthreads();
  if (f < 8)  red[f] += red[f + 8];  __syncthreads();
  if (f < 4)  red[f] += red[f + 4];  __syncthreads();
  if (f < 2)  red[f] += red[f + 2];  __syncthreads();
  __syncthreads();
  const float mean = (red[0] + red[1]) * (1.f / DF);
  __syncthreads();
  const float d = v - mean;
  red[f] = d * d; __syncthreads();
  if (f < 64) red[f] += red[f + 64]; __syncthreads();
  if (f < 32) red[f] += red[f + 32]; __syncthreads();
  if (f < 16) red[f] += red[f + 16]; __syncthreads();
  if (f < 8)  red[f] += red[f + 8];  __syncthreads();
  if (f < 4)  red[f] += red[f + 4];  __syncthreads();
  if (f < 2)  red[f] += red[f + 2];  __syncthreads();
  __syncthreads();
  const float var = (red[0] + red[1]) * (1.f / DF);
  dst[(size_t)r * DF + f] = gamma[f] * d * rsqrtf(var + 1e-5f) + beta[f];
}

// ---------------------------------------------------------------- driver
extern "C" void kernel_launch(void* const* d_in, const int* in_sizes, int n_in,
                              void* d_out, int out_size, void* d_ws, size_t ws_size,
                              hipStream_t stream) {
  const int N  = in_sizes[0] / DF;
  const int Ew = in_sizes[6] / 2;
  const int Et = in_sizes[7] / 2;
  const float* x_wave   = (const float*)d_in[0];
  const float* x_trans  = (const float*)d_in[1];
  const float* x_target = (const float*)d_in[2];
  const float* t_wave   = (const float*)d_in[3];
  const float* t_trans  = (const float*)d_in[4];
  const float* t_target = (const float*)d_in[5];
  const int*   e_wt     = (const int*)d_in[6];
  const int*   e_tt     = (const int*)d_in[7];
  const float* gamma    = (const float*)d_in[44];
  const float* beta     = (const float*)d_in[45];
  const float* eap      = (const float*)d_in[46];
  const float* rwp      = (const float*)d_in[47];
  float* out = (float*)d_out;

  const size_t ND = (size_t)N * DF;
  float* ws = (float*)d_ws;
  float* QE = ws;                 // q_w result; reused for x_out
  float* KE = QE + ND;            // k_w result; reused for t_out
  float* A1 = KE + ND;            // x_src @ mlp1_a
  float* B1 = A1 + ND;            // x_tgt @ mlp1_b + b1
  float* Qm = B1 + ND;            // x_tgt @ Wq
  float* Km = Qm + ND;            // x_src @ Wk
  float* Vb = Km + ND;            // x_src @ Wv
  float* WV = Vb + ND;            // sum(a*v) per target
  float* TS = WV + ND;            // sum(t_src rows) per target
  const size_t E4m = (size_t)((Ew > Et) ? Ew : Et) * 4;
  float* buf1 = TS + ND;          // comb -> exp -> p
  float* buf2 = buf1 + E4m;       // scores -> weighted
  float* nmax = buf2 + E4m;       // [N,4]
  float* nden = nmax + (size_t)N * 4;  // [N,4] (deg follows, zeroed together)
  float* deg  = nden + (size_t)N * 4;  // [N]
  float* g8   = deg + N;          // gmax[4], gsum[4]

  auto gb = [](size_t n) { return dim3((unsigned)((n + 255) / 256)); };
  const dim3 gg((unsigned)((N + 31) / 32)), gblk(256);   // 32 rows per block

  auto conv = [&](const float* xs, const float* xt, const float* tsrc,
                  const int* eidx, int E_, int pb,
                  const float* xprev, float* xdst,
                  const float* tprev, float* tdst) {
    const int* sidx = eidx;
    const int* tidx = eidx + E_;
    const float* Wq = (const float*)d_in[pb + 0];
    const float* Wk = (const float*)d_in[pb + 1];
    const float* Wv = (const float*)d_in[pb + 2];
    const float* Wt = (const float*)d_in[pb + 3];
    const float* Wow = (const float*)d_in[pb + 4];
    const float* Wob = (const float*)d_in[pb + 5];
    const float* m1w = (const float*)d_in[pb + 6];
    const float* m1b = (const float*)d_in[pb + 7];
    const float* m2w = (const float*)d_in[pb + 8];
    const float* m2b = (const float*)d_in[pb + 9];
    const float* qw = (const float*)d_in[pb + 10];
    const float* qb = (const float*)d_in[pb + 11];
    const float* kw = (const float*)d_in[pb + 12];
    const float* kb = (const float*)d_in[pb + 13];
    const float* swp = (const float*)d_in[pb + 16];
    const float* fwp = (const float*)d_in[pb + 17];
    const int tot4 = E_ * 4;

    fillk<<<gb(2 * ND), 256, 0, stream>>>(WV, 0.f, 2 * ND);            // WV + TS
    fillk<<<gb((size_t)N * 4), 256, 0, stream>>>(nmax, -INFINITY, (size_t)N * 4);
    fillk<<<gb((size_t)N * 5), 256, 0, stream>>>(nden, 0.f, (size_t)N * 5);  // nden+deg
    fillk<<<1, 4, 0, stream>>>(g8, -INFINITY, 4);
    fillk<<<1, 4, 0, stream>>>(g8 + 4, 0.f, 4);

    gemm_xwt<1><<<gg, gblk, 0, stream>>>(xs, qw, qb, nullptr, QE, N, DF);
    gemm_xwt<1><<<gg, gblk, 0, stream>>>(xt, kw, kb, nullptr, KE, N, DF);
    gemm_xwt<0><<<gg, gblk, 0, stream>>>(xs, m1w, nullptr, nullptr, A1, N, 2 * DF);
    gemm_xwt<1><<<gg, gblk, 0, stream>>>(xt, m1w + DF, m1b, nullptr, B1, N, 2 * DF);
    gemm_xwt<0><<<gg, gblk, 0, stream>>>(xt, Wq, nullptr, nullptr, Qm, N, DF);
    gemm_xwt<0><<<gg, gblk, 0, stream>>>(xs, Wk, nullptr, nullptr, Km, N, DF);
    gemm_xwt<0><<<gg, gblk, 0, stream>>>(xs, Wv, nullptr, nullptr, Vb, N, DF);

    edge_pass1<<<E_, 128, 0, stream>>>(sidx, tidx, A1, B1, QE, KE, Qm, Km,
                                       m2w, m2b, swp, fwp, buf1, buf2);
    reduce_max4<<<1024, 256, 0, stream>>>(buf1, g8, tot4);
    edge_exp_sum<<<1024, 256, 0, stream>>>(buf1, g8, g8 + 4, tot4);
    edge_weighted<<<(tot4 + 255) / 256, 256, 0, stream>>>(tidx, g8 + 4, buf1, buf2,
                                                          nmax, tot4);
    edge_expnode<<<(tot4 + 255) / 256, 256, 0, stream>>>(tidx, nmax, buf2, buf1,
                                                         nden, tot4);
    edge_scatter<<<E_, 128, 0, stream>>>(sidx, tidx, buf1, nden, Vb, tsrc, WV, TS, deg);

    gemm_xwt<2><<<gg, gblk, 0, stream>>>(WV, Wow, Wob, deg, QE, N, DF);      // x_out
    gemm_xwt<0><<<gg, gblk, 0, stream>>>(TS, Wt, nullptr, nullptr, KE, N, DF); // t_out

    epilogue_ln<<<N, 128, 0, stream>>>(xprev, QE, rwp, eap, gamma, beta, xdst);
    epilogue_ln<<<N, 128, 0, stream>>>(tprev, KE, rwp, eap, gamma, beta, tdst);
  };

  // conv1: wave -> trans,   outputs x_tr (seg0) and t_tr (seg2)
  conv(x_wave, x_trans, t_wave, e_wt, Ew, 8,
       x_trans, out + 0 * ND, t_trans, out + 2 * ND);
  // conv2: trans -> target, outputs x_tg (seg1) and t_tg (seg3)
  conv(out + 0 * ND, x_target, out + 2 * ND, e_tt, Et, 26,
       x_target, out + 1 * ND, t_target, out + 3 * ND);
}